// InducGEN_70540542870036
// MI455X (gfx1250) — compile-verified
//
#include <hip/hip_runtime.h>
#include <hip/hip_bf16.h>

typedef float v2f __attribute__((ext_vector_type(2)));
typedef float v8f __attribute__((ext_vector_type(8)));

#define DE_DIM   200           // entity / out dim
#define DR_DIM   200           // relation dim
#define K_DIM    400           // DE+DR
#define NB       4             // bases
#define N_DIM    (NB * DE_DIM) // 800 = GEMM N
#define N_TILES  (N_DIM / 16)  // 50, exact
#define TILE_M   16
#define MAX_F    4096          // cap on filtered edges
#define MAX_TILES (MAX_F / TILE_M)

struct Ws {
    int   count;        // number of edges with dst == unseen
    int   pad[3];
    float acc[256];     // output accumulator (200 used)
    int   list[MAX_F];  // compacted edge ids
};

__global__ void induc_init_kernel(Ws* ws) {
    int i = threadIdx.x;
    if (i == 0) ws->count = 0;
    if (i < 256) ws->acc[i] = 0.0f;
}

__global__ void induc_filter_kernel(const int* __restrict__ edge_dst,
                                    const int* __restrict__ unseen_p,
                                    int num_edges, Ws* ws) {
    int e = blockIdx.x * blockDim.x + threadIdx.x;
    if (e >= num_edges) return;
    if (edge_dst[e] == unseen_p[0]) {
        int pos = atomicAdd(&ws->count, 1);
        if (pos < MAX_F) ws->list[pos] = e;
    }
}

// One block per tile of 16 filtered edges. Wave 0 runs the fp32 WMMA GEMM
// h[16, 800] = A[16, 400] x basis[400, 800]; epilogue folds in att[edge_type]
// and accumulates into ws->acc (all surviving edges share the same dst node).
__global__ void induc_gemm_kernel(const float* __restrict__ ent,
                                  const float* __restrict__ relemb,
                                  const float* __restrict__ basis,
                                  const float* __restrict__ att,
                                  const int*   __restrict__ node_id,
                                  const int*   __restrict__ edge_src,
                                  const int*   __restrict__ edge_type,
                                  const int*   __restrict__ rel_index,
                                  Ws* ws) {
    __shared__ float Atile[TILE_M][K_DIM + 4];

    int cnt   = ws->count;
    int nproc = cnt < MAX_F ? cnt : MAX_F;
    int tbase = blockIdx.x * TILE_M;
    if (tbase >= nproc) return;
    int valid = nproc - tbase;
    if (valid > TILE_M) valid = TILE_M;

    int tid = threadIdx.x;
    // Stage A tile: row m = filtered edge, col k = concat(x[src], rel_emb)
    for (int idx = tid; idx < TILE_M * K_DIM; idx += blockDim.x) {
        int m = idx / K_DIM;
        int k = idx - m * K_DIM;
        float v = 0.0f;
        if (m < valid) {
            int e = ws->list[tbase + m];
            if (k < DE_DIM) {
                int n = edge_src[e];
                v = ent[(size_t)node_id[n] * DE_DIM + k];
            } else {
                v = relemb[(size_t)rel_index[e] * DR_DIM + (k - DE_DIM)];
            }
        }
        Atile[m][k] = v;
    }
    __syncthreads();

    if (tid < 32) {                       // wave 0, EXEC all-1s (wave32)
        int lane = tid;
        int mrow = lane & 15;             // A: M index
        int koff = (lane >> 4) << 1;      // A/B: K sub-offset 0 or 2
        int ncol = lane & 15;             // B/C: N index within tile
        int mbase = (lane >> 4) << 3;     // C: M base 0 or 8

        for (int nt = 0; nt < N_TILES; ++nt) {
            int col = nt * 16 + ncol;     // global N column in [0, 800)
            int bb  = col / DE_DIM;       // basis index
            int oo  = col - bb * DE_DIM;  // output channel
            const float* bcol = basis + (size_t)bb * K_DIM * DE_DIM + oo;

            v8f c = {};
            for (int kb = 0; kb < K_DIM; kb += 4) {
                v2f a, b;
                a.x = Atile[mrow][kb + koff];
                a.y = Atile[mrow][kb + koff + 1];
                b.x = bcol[(size_t)(kb + koff)     * DE_DIM];
                b.y = bcol[(size_t)(kb + koff + 1) * DE_DIM];
                c = __builtin_amdgcn_wmma_f32_16x16x4_f32(
                        /*neg_a=*/false, a, /*neg_b=*/false, b,
                        /*c_mod=*/(short)0, c,
                        /*reuse_a=*/false, /*reuse_b=*/false);
            }
            // Epilogue: msg[e][oo] += att[type(e)][bb] * h; same dst node,
            // so accumulate straight into the shared 200-float accumulator.
            #pragma unroll
            for (int r = 0; r < 8; ++r) {
                int m = mbase + r;
                if (m < valid) {
                    int e = ws->list[tbase + m];
                    int t = edge_type[e];
                    float w = att[(size_t)t * NB + bb];
                    atomicAdd(&ws->acc[oo], w * c[r]);
                }
            }
        }
    }
}

__global__ void induc_finalize_kernel(const Ws* __restrict__ ws,
                                      float* __restrict__ out) {
    int i = threadIdx.x;
    if (i < DE_DIM) {
        float c = (float)ws->count;       // = cnt[unseen] in the reference
        out[i] = ws->acc[i] / fmaxf(c, 1.0f);
    }
}

extern "C" void kernel_launch(void* const* d_in, const int* in_sizes, int n_in,
                              void* d_out, int out_size, void* d_ws, size_t ws_size,
                              hipStream_t stream) {
    const float* ent     = (const float*)d_in[0];
    const float* relemb  = (const float*)d_in[1];
    const float* basis   = (const float*)d_in[2];
    const float* att     = (const float*)d_in[3];
    const int*   node_id = (const int*)d_in[4];
    const int*   e_src   = (const int*)d_in[5];
    const int*   e_dst   = (const int*)d_in[6];
    const int*   e_type  = (const int*)d_in[7];
    const int*   r_idx   = (const int*)d_in[8];
    const int*   unseen  = (const int*)d_in[9];
    float* out = (float*)d_out;
    Ws*    ws  = (Ws*)d_ws;

    int num_edges = in_sizes[5];

    induc_init_kernel<<<1, 256, 0, stream>>>(ws);
    induc_filter_kernel<<<(num_edges + 255) / 256, 256, 0, stream>>>(
        e_dst, unseen, num_edges, ws);
    induc_gemm_kernel<<<MAX_TILES, 256, 0, stream>>>(
        ent, relemb, basis, att, node_id, e_src, e_type, r_idx, ws);
    induc_finalize_kernel<<<1, 256, 0, stream>>>(ws, out);
}